// ApproximationLayer_24163486007473
// MI455X (gfx1250) — compile-verified
//
#include <hip/hip_runtime.h>
#include <stdint.h>

// ---------------------------------------------------------------------------
// Problem: out = x, except out[b, rows[i], cols[j]] = mute_msb(x[b, rows[i], cols[j]])
//   B=256, R=C=1024, 64x64 indexed grid. Memory-bound: ~2.15 GB traffic,
//   floor ~92us @ 23.3 TB/s. Strategy: NT-streamed b128 copy + tiny async-LDS
//   gather/transform/scatter kernel for the indexed corner.
// ---------------------------------------------------------------------------

typedef float v4f __attribute__((ext_vector_type(4)));

#define AS_GLOBAL __attribute__((address_space(1)))
#define AS_LDS    __attribute__((address_space(3)))

// mute_msb: x = m * 2^e (frexp, m in [0.5,1)). If e > 0 (i.e. biased exp >= 127,
// excluding inf/nan), result is m: rebias exponent to 126. Else identity.
__device__ __forceinline__ float mute_msb(float x) {
  uint32_t u = __float_as_uint(x);
  uint32_t be = (u >> 23) & 0xFFu;
  uint32_t muted = (u & 0x807FFFFFu) | (126u << 23);
  return (be >= 127u && be != 255u) ? __uint_as_float(muted) : x;
}

// --- CDNA5 async global->LDS copy (ASYNCcnt path), with asm fallback --------
// Probe round 1 showed the builtin exists with signature taking
// (global int*, shared int*, imm offset, imm cpol) — clang printed the param
// as "__device__ int *" (HIP pretty-print of addrspace(1) int*).
__device__ __forceinline__ void async_g2l_b32(const float* g, float* l) {
#if defined(__HIP_DEVICE_COMPILE__)
#if __has_builtin(__builtin_amdgcn_global_load_async_to_lds_b32)
  __builtin_amdgcn_global_load_async_to_lds_b32(
      (AS_GLOBAL int*)(uintptr_t)g, (AS_LDS int*)l, /*offset=*/0, /*cpol=*/0);
#else
  unsigned lds_addr = (unsigned)(uintptr_t)(AS_LDS float*)l;
  unsigned long long gaddr = (unsigned long long)(uintptr_t)g;
  asm volatile("global_load_async_to_lds_b32 %0, %1, off"
               :: "v"(lds_addr), "v"(gaddr)
               : "memory");
#endif
#else
  (void)g; (void)l; // host-side parse only; never executed
#endif
}

__device__ __forceinline__ void wait_asynccnt0() {
#if defined(__HIP_DEVICE_COMPILE__)
#if __has_builtin(__builtin_amdgcn_s_wait_asynccnt)
  __builtin_amdgcn_s_wait_asynccnt(0);
#else
  asm volatile("s_wait_asynccnt 0" ::: "memory");
#endif
#endif
}

// --- Kernel 1: full-tensor streaming copy, non-temporal b128 both ways ------
__global__ __launch_bounds__(256) void bulk_copy_kernel(
    const v4f* __restrict__ in, v4f* __restrict__ out, size_t n4) {
  const size_t stride = (size_t)gridDim.x * blockDim.x;
  const size_t t = (size_t)blockIdx.x * blockDim.x + threadIdx.x;

  size_t i0 = t;
  size_t i1 = t + stride;
  size_t i2 = t + 2 * stride;
  size_t i3 = t + 3 * stride;

  if (i3 < n4) {
    // 4 loads in flight, then 4 stores: good MLP, perfect per-instr coalescing.
    v4f a = __builtin_nontemporal_load(in + i0);
    v4f b = __builtin_nontemporal_load(in + i1);
    v4f c = __builtin_nontemporal_load(in + i2);
    v4f d = __builtin_nontemporal_load(in + i3);
    __builtin_nontemporal_store(a, out + i0);
    __builtin_nontemporal_store(b, out + i1);
    __builtin_nontemporal_store(c, out + i2);
    __builtin_nontemporal_store(d, out + i3);
  } else {
    for (size_t i = i0; i < n4; i += stride) {
      __builtin_nontemporal_store(__builtin_nontemporal_load(in + i), out + i);
    }
  }
}

// --- Kernel 2: indexed corner. One block per batch. --------------------------
// Gather x[b, rows[i], cols[j]] through the async global->LDS path, wait on
// ASYNCcnt, transform in VALU, scatter to out. Each lane reads only LDS slots
// it itself async-loaded, so per-wave s_wait_asynccnt 0 is sufficient (no
// workgroup barrier needed).
#define CORNER_TILE 4096

__global__ __launch_bounds__(256) void corner_kernel(
    const float* __restrict__ x, const int* __restrict__ rows,
    const int* __restrict__ cols, float* __restrict__ out,
    int n_rows, int n_cols, int R, int C) {
  __shared__ float tile[CORNER_TILE]; // 16 KB of the 320 KB WGP LDS

  const size_t batch = (size_t)blockIdx.x * (size_t)R * (size_t)C;
  const int total = n_rows * n_cols;

  if (total <= CORNER_TILE) {
    for (int k = threadIdx.x; k < total; k += blockDim.x) {
      const int i = k / n_cols;
      const int j = k - i * n_cols;
      const size_t off = batch + (size_t)rows[i] * (size_t)C + (size_t)cols[j];
      async_g2l_b32(x + off, &tile[k]);
    }
    wait_asynccnt0();
    for (int k = threadIdx.x; k < total; k += blockDim.x) {
      const int i = k / n_cols;
      const int j = k - i * n_cols;
      const size_t off = batch + (size_t)rows[i] * (size_t)C + (size_t)cols[j];
      out[off] = mute_msb(tile[k]);
    }
  } else {
    // Defensive fallback for larger index sets: direct path, no LDS staging.
    for (int k = threadIdx.x; k < total; k += blockDim.x) {
      const int i = k / n_cols;
      const int j = k - i * n_cols;
      const size_t off = batch + (size_t)rows[i] * (size_t)C + (size_t)cols[j];
      out[off] = mute_msb(x[off]);
    }
  }
}

// ---------------------------------------------------------------------------
extern "C" void kernel_launch(void* const* d_in, const int* in_sizes, int n_in,
                              void* d_out, int out_size, void* d_ws, size_t ws_size,
                              hipStream_t stream) {
  (void)n_in; (void)d_ws; (void)ws_size;

  const float* x   = (const float*)d_in[0];
  const int* rows  = (const int*)d_in[1];
  const int* cols  = (const int*)d_in[2];
  float* out       = (float*)d_out;

  const int R = 1024, C = 1024;
  const int n_rows = in_sizes[1];
  const int n_cols = in_sizes[2];
  const int B = in_sizes[0] / (R * C);

  // ---- Pass 1: full streaming copy (roofline path) ----
  const size_t n4 = (size_t)out_size / 4; // float4 chunks; out_size = B*R*C
  const int threads = 256;
  const int unroll = 4;
  size_t total_threads = (n4 + unroll - 1) / unroll;
  unsigned blocks = (unsigned)((total_threads + threads - 1) / threads);
  bulk_copy_kernel<<<blocks, threads, 0, stream>>>(
      (const v4f*)x, (v4f*)out, n4);

  // ---- Pass 2: overwrite the indexed grid with the muted values ----
  corner_kernel<<<B, threads, 0, stream>>>(x, rows, cols, out,
                                           n_rows, n_cols, R, C);
}